// SNNLayer_90288802496554
// MI455X (gfx1250) — compile-verified
//
#include <hip/hip_runtime.h>
#include <hip/hip_bf16.h>
#include <stdint.h>

// SNN layer: y[b,o,t] = LIF-scan over i[t,b,o] where i = einsum('bit,io->tbo', x, w)
// B=64, n_in=512, n_out=512, T=500. fp32 WMMA (V_WMMA_F32_16X16X4_F32): w has
// std=8000 and the scan thresholds at 1.0, so low-precision matrix paths are unsafe.
// A-operand tiles are staged LDS-side with the Tensor Data Mover (tensor_load_to_lds),
// so each x element is fetched from memory exactly once device-wide.

#define B_DIM   64
#define NIN     512
#define NOUT    512
#define T_DIM   500
#define M_DIM   (B_DIM * T_DIM)      // 32000 logical rows
#define ALPHA   (0.001f / 50.0f)     // MS / TAU
#define V_TH    1.0f

#define IWS_FLOATS ((size_t)M_DIM * NOUT)   // 65.5 MB intermediate (L2-resident)

typedef float        v2f __attribute__((ext_vector_type(2)));
typedef float        v8f __attribute__((ext_vector_type(8)));
typedef unsigned int v4u __attribute__((ext_vector_type(4)));
typedef int          v4i __attribute__((ext_vector_type(4)));
typedef int          v8i __attribute__((ext_vector_type(8)));

// ---------------------------------------------------------------------------
// Kernel 0: pack w into K-pair-interleaved layout so a B fragment is one b64:
//   wpk[(p*NOUT + n)*2 + j] = w[(2p+j)*NOUT + n],  p = 0..255
// ---------------------------------------------------------------------------
__global__ __launch_bounds__(256)
void snn_pack_w(const float* __restrict__ w, float* __restrict__ wpk) {
    const int g = blockIdx.x * 256 + threadIdx.x;   // 0 .. 131071
    const int p = g >> 9;          // k-pair index
    const int n = g & 511;
    float2 pr;
    pr.x = w[(size_t)(2 * p) * NOUT + n];
    pr.y = w[(size_t)(2 * p + 1) * NOUT + n];
    *(float2*)(wpk + (size_t)g * 2) = pr;           // coalesced b64 store
}

// ---------------------------------------------------------------------------
// Kernel 1: i_ws[(b*T + t)*NOUT + o] = sum_k x[b,k,t] * w[k,o]
// Grid: (64 b, 16 t-slabs) x 256 threads (8 waves). Block owns a 32-t slab
// (two 16-row M tiles); wave wv owns N columns [wv*64, wv*64+64).
// TDM stages x[b, 0..511, t0..t0+31] into LDS (zeros past t=499); per K-step:
// 4 LDS dwords (A) + 4 global b64 (B) vs 8 WMMA.
// ---------------------------------------------------------------------------
__global__ __launch_bounds__(256)
void snn_gemm_wmma(const float* __restrict__ x,
                   const float* __restrict__ wpk,
                   float* __restrict__ i_ws) {
    extern __shared__ float sA[];         // [k][tt] = [512][32] floats = 64 KB

    const int tid  = threadIdx.x;
    const int wave = tid >> 5;            // 0..7
    const int lane = tid & 31;
    const int l16  = lane & 15;
    const int h    = lane >> 4;           // 0/1: K-half selector for A/B frags
    const int b    = blockIdx.x;          // 0..63
    const int t0   = blockIdx.y * 32;     // 0,32,...,480

    // ---- TDM: async DMA of the A tile into LDS (wave 0 only) ----
    if (wave == 0) {
        const uint64_t ga  = (uint64_t)(uintptr_t)(x + (size_t)b * NIN * T_DIM + t0);
        const uint32_t lds = (uint32_t)(uintptr_t)(void*)&sA[0];   // LDS byte offset

        v4u g0;
        g0[0] = 1u;                                   // count=1 (valid user D#)
        g0[1] = lds;                                  // lds_addr
        g0[2] = (uint32_t)ga;                         // global_addr[31:0]
        g0[3] = ((uint32_t)(ga >> 32) & 0x01FFFFFFu)  // global_addr[56:32]
              | (2u << 30);                           // type = 2 ("image")

        v8i g1;
        g1[0] = (int)(2u << 16);                      // data_size = 2 -> 4 bytes
        g1[1] = (int)((uint32_t)T_DIM << 16);         // tensor_dim0 = 500 (t)
        g1[2] = (int)((uint32_t)NIN << 16);           // tensor_dim1 = 512 (k)
        g1[3] = (int)(32u << 16);                     // tile_dim0 = 32 (t)
        g1[4] = (int)NIN;                             // tile_dim1 = 512 (k)
        g1[5] = (int)T_DIM;                           // tensor_dim0_stride = 500
        g1[6] = 0;                                    // dim1 stride unused (2D)
        g1[7] = 0;

        v4i gz4; gz4[0] = 0; gz4[1] = 0; gz4[2] = 0; gz4[3] = 0;  // 2D: groups 2/3 zero
        v8i gz8 = {};                                             // unused (6-arg form)

        __builtin_amdgcn_tensor_load_to_lds(g0, g1, gz4, gz4, gz8, 0);
        __builtin_amdgcn_s_wait_tensorcnt(0);
    }
    __syncthreads();

    const int n_base = wave * 64;
    // Lane's B-fragment base in packed w: pair index (k0/2 + h), column n
    const float* wp = wpk + (size_t)(n_base + l16) * 2;

    v8f a0n0 = {}, a0n1 = {}, a0n2 = {}, a0n3 = {};
    v8f a1n0 = {}, a1n1 = {}, a1n2 = {}, a1n3 = {};

    #pragma unroll 2
    for (int k0 = 0; k0 < NIN; k0 += 4) {
        const int ka = k0 + 2 * h;        // this lane's K pair: ka, ka+1

        // A fragments from LDS: sA[k*32 + tt]; pairs 128B apart -> ds_load_2addr
        v2f af0, af1;
        af0.x = sA[ka * 32 + l16];
        af0.y = sA[(ka + 1) * 32 + l16];
        af1.x = sA[ka * 32 + 16 + l16];
        af1.y = sA[(ka + 1) * 32 + 16 + l16];

        // B fragments: one 8-byte load each from packed w (L2-resident, 1 MB)
        const float* wrow = wp + (size_t)(k0 / 2 + h) * (NOUT * 2);
        const v2f bf0 = *(const v2f*)(wrow);
        const v2f bf1 = *(const v2f*)(wrow + 32);    // +16 columns * 2
        const v2f bf2 = *(const v2f*)(wrow + 64);
        const v2f bf3 = *(const v2f*)(wrow + 96);

        a0n0 = __builtin_amdgcn_wmma_f32_16x16x4_f32(false, af0, false, bf0, (short)0, a0n0, false, false);
        a1n0 = __builtin_amdgcn_wmma_f32_16x16x4_f32(false, af1, false, bf0, (short)0, a1n0, false, false);
        a0n1 = __builtin_amdgcn_wmma_f32_16x16x4_f32(false, af0, false, bf1, (short)0, a0n1, false, false);
        a1n1 = __builtin_amdgcn_wmma_f32_16x16x4_f32(false, af1, false, bf1, (short)0, a1n1, false, false);
        a0n2 = __builtin_amdgcn_wmma_f32_16x16x4_f32(false, af0, false, bf2, (short)0, a0n2, false, false);
        a1n2 = __builtin_amdgcn_wmma_f32_16x16x4_f32(false, af1, false, bf2, (short)0, a1n2, false, false);
        a0n3 = __builtin_amdgcn_wmma_f32_16x16x4_f32(false, af0, false, bf3, (short)0, a0n3, false, false);
        a1n3 = __builtin_amdgcn_wmma_f32_16x16x4_f32(false, af1, false, bf3, (short)0, a1n3, false, false);
    }

    // C/D layout: VGPR r -> (M = r + 8*h, N = lane&15). Store i as [b][t][o];
    // mask rows with t >= 500 (zero-padded tail of the last slab).
    #pragma unroll
    for (int r = 0; r < 8; ++r) {
        const int tr0 = t0 + r + 8 * h;       // m-tile 0 row
        const int tr1 = tr0 + 16;             // m-tile 1 row
        if (tr0 < T_DIM) {
            const size_t row = ((size_t)b * T_DIM + tr0) * NOUT + n_base + l16;
            i_ws[row +  0] = a0n0[r];
            i_ws[row + 16] = a0n1[r];
            i_ws[row + 32] = a0n2[r];
            i_ws[row + 48] = a0n3[r];
        }
        if (tr1 < T_DIM) {
            const size_t row = ((size_t)b * T_DIM + tr1) * NOUT + n_base + l16;
            i_ws[row +  0] = a1n0[r];
            i_ws[row + 16] = a1n1[r];
            i_ws[row + 32] = a1n2[r];
            i_ws[row + 48] = a1n3[r];
        }
    }
}

// ---------------------------------------------------------------------------
// Kernel 2: per-(b,o) LIF scan over t. Reads i_ws[(b*T+t)*NOUT + o] (o is the
// fast lane index -> 128B/wave coalesced, L2-resident). Buffers 16 spikes in
// registers, writes y[b,o,t] as 4 aligned float4 stores (contiguous 64B/thread).
// ---------------------------------------------------------------------------
__global__ __launch_bounds__(256)
void snn_scan(const float* __restrict__ i_ws,
              float* __restrict__ y) {
    const int b = blockIdx.x;                       // 0..63
    const int o = blockIdx.y * 256 + threadIdx.x;   // 0..511

    const float* ib = i_ws + (size_t)b * T_DIM * NOUT + o;   // step t: += NOUT
    float* yb = y + ((size_t)b * NOUT + o) * T_DIM;          // contiguous in t

    float v = 0.0f;
    float buf[16];

    for (int c = 0; c < 31; ++c) {                  // t = 0..495
        const int t0 = c * 16;
        #pragma unroll
        for (int j = 0; j < 16; ++j) {
            const float it = ib[(size_t)(t0 + j) * NOUT];
            v = v + (it - v) * ALPHA;
            const float spike = (v >= V_TH) ? 1.0f : 0.0f;
            v = (v >= V_TH) ? 0.0f : v;
            buf[j] = spike;
        }
        float4* dst = (float4*)(yb + t0);           // 16B-aligned (stride 2000B)
        dst[0] = make_float4(buf[0],  buf[1],  buf[2],  buf[3]);
        dst[1] = make_float4(buf[4],  buf[5],  buf[6],  buf[7]);
        dst[2] = make_float4(buf[8],  buf[9],  buf[10], buf[11]);
        dst[3] = make_float4(buf[12], buf[13], buf[14], buf[15]);
    }

    {                                               // tail: t = 496..499
        #pragma unroll
        for (int j = 0; j < 4; ++j) {
            const float it = ib[(size_t)(496 + j) * NOUT];
            v = v + (it - v) * ALPHA;
            const float spike = (v >= V_TH) ? 1.0f : 0.0f;
            v = (v >= V_TH) ? 0.0f : v;
            buf[j] = spike;
        }
        float4* dst = (float4*)(yb + 496);
        dst[0] = make_float4(buf[0], buf[1], buf[2], buf[3]);
    }
}

extern "C" void kernel_launch(void* const* d_in, const int* in_sizes, int n_in,
                              void* d_out, int out_size, void* d_ws, size_t ws_size,
                              hipStream_t stream) {
    const float* x = (const float*)d_in[0];   // [64, 512, 500]
    const float* w = (const float*)d_in[1];   // [512, 512]
    float* y    = (float*)d_out;              // [64, 512, 500]
    float* i_ws = (float*)d_ws;               // [32000, 512] = 65.5 MB
    float* wpk  = i_ws + IWS_FLOATS;          // packed w, 1 MB (8B-aligned offset)

    (void)in_sizes; (void)n_in; (void)out_size; (void)ws_size;

    snn_pack_w<<<dim3((NIN / 2) * NOUT / 256), dim3(256), 0, stream>>>(w, wpk);
    snn_gemm_wmma<<<dim3(B_DIM, 16), dim3(256), NIN * 32 * sizeof(float), stream>>>(x, wpk, i_ws);
    snn_scan<<<dim3(B_DIM, NOUT / 256), dim3(256), 0, stream>>>(i_ws, y);
}